// LightGCN_66185446031939
// MI455X (gfx1250) — compile-verified
//
#include <hip/hip_runtime.h>

// ---------------- problem constants (from reference) ----------------
#define NN     50000
#define DD     64
#define EE     800000
#define P_DROP 0.1f

typedef __attribute__((ext_vector_type(2))) float v2f;
typedef __attribute__((ext_vector_type(8))) float v8f;

// stateless per-element dropout decision: depends only on (element, seed),
// so it is consistent across every edge that gathers the same source row.
__device__ __forceinline__ float drop_mask(unsigned idx, unsigned seed)
{
    unsigned h = idx * 0x9E3779B9u + seed;
    h ^= h >> 16; h *= 0x7feb352du;
    h ^= h >> 15; h *= 0x846ca68bu;
    h ^= h >> 16;
    const unsigned thresh = (unsigned)(P_DROP * 4294967296.0);   // drop if h < thresh
    const float scale = 1.0f / (1.0f - P_DROP);
    return (h >= thresh) ? scale : 0.0f;
}

// ---------------- h0 = embed_weight[x]  (float4-vectorized gather) --------
__global__ __launch_bounds__(256) void gather_embed_kernel(
    const int* __restrict__ x, const float* __restrict__ w, float* __restrict__ h0)
{
    int t = blockIdx.x * 256 + threadIdx.x;        // over N * (D/4) float4 groups
    if (t >= NN * (DD / 4)) return;
    int row = t >> 4;                              // D/4 == 16
    int c4  = t & 15;
    int src = x[row];
    reinterpret_cast<float4*>(h0)[(long)row * 16 + c4] =
        reinterpret_cast<const float4*>(w)[(long)src * 16 + c4];
}

// ---------------- zero the three scatter targets --------------------------
__global__ __launch_bounds__(256) void zero_kernel(float4* __restrict__ p, int n4)
{
    int t = blockIdx.x * 256 + threadIdx.x;
    if (t < n4) p[t] = make_float4(0.f, 0.f, 0.f, 0.f);
}

// ---------------- fused dropout + SpMM ------------------------------------
// 16 lanes per edge; each lane: one B128 gather of h[src], 4 inline dropout
// hashes (pure VALU, free under the atomic/L2 bound), 4 HW f32 atomics.
__global__ __launch_bounds__(256) void spmm_dropout_kernel(
    const int*   __restrict__ esrc, const int*   __restrict__ edst,
    const float* __restrict__ ew,   const float* __restrict__ hin,
    float*       __restrict__ hout, unsigned seed)
{
    long t = (long)blockIdx.x * 256 + threadIdx.x;
    int  e = (int)(t >> 4);
    if (e >= EE) return;
    int  c = ((int)t & 15) << 2;                   // feature offset 0..60 step 4

    // stream-prefetch edge arrays far ahead (global_prefetch_b8)
    if (((int)t & 15) == 0 && e + 4096 < EE) {
        __builtin_prefetch(esrc + e + 4096, 0, 1);
        __builtin_prefetch(edst + e + 4096, 0, 1);
        __builtin_prefetch(ew   + e + 4096, 0, 1);
    }

    int   s = esrc[e];
    int   d = edst[e];
    float w = ew[e];

    float4 v = *reinterpret_cast<const float4*>(hin + (long)s * DD + c);
    unsigned base = (unsigned)(s * DD + c);
    v.x *= drop_mask(base + 0u, seed);
    v.y *= drop_mask(base + 1u, seed);
    v.z *= drop_mask(base + 2u, seed);
    v.w *= drop_mask(base + 3u, seed);

    float* o = hout + (long)d * DD + c;
    unsafeAtomicAdd(o + 0, w * v.x);               // global_atomic_add_f32
    unsafeAtomicAdd(o + 1, w * v.y);
    unsafeAtomicAdd(o + 2, w * v.z);
    unsafeAtomicAdd(o + 3, w * v.w);
}

// ---------------- out = 0.25*(h0+h1+h2+h3) via exact identity WMMA --------
// One wave32 per 16x16 tile. For each layer L:
//   out_tile += (0.25*I16) x H_L_tile, split into four K=4 slices of
//   V_WMMA_F32_16X16X4_F32, chained through the accumulator.
// A (16x4 f32, 2 VGPRs): lane m=lane&15; VGPR0 holds K={0|2}, VGPR1 K={1|3}
//   selected by lane-half (ISA 7.12.2 "32-bit A-Matrix 16x4").
// B (4x16 f32, 2 VGPRs): lane n=lane&15; same K split across lane halves.
// C/D (16x16 f32, 8 VGPRs): VGPR r -> M = r + (lane>=16 ? 8 : 0), N = lane&15.
__global__ __launch_bounds__(128) void combine_wmma_kernel(
    const float* __restrict__ h0, const float* __restrict__ h1,
    const float* __restrict__ h2, const float* __restrict__ h3,
    float* __restrict__ out)
{
    int wave = blockIdx.x * 4 + (threadIdx.x >> 5);
    const int ntiles = (NN / 16) * (DD / 16);      // 3125 * 4 = 12500
    if (wave >= ntiles) return;

    int  rb   = wave >> 2;                         // 16-node row block
    int  cb   = wave & 3;                          // 16-feature col block
    long base = (long)rb * 16 * DD + cb * 16;

    int lane = threadIdx.x & 31;
    int lm   = lane & 15;
    int hi   = lane >> 4;

    v8f acc = {};

    const float* bufs[4] = { h0, h1, h2, h3 };
#pragma unroll
    for (int L = 0; L < 4; ++L) {
        const float* H = bufs[L] + base;
#pragma unroll
        for (int j = 0; j < 4; ++j) {              // K slices of the identity
            int k0 = 4 * j + (hi ? 2 : 0);
            int k1 = 4 * j + (hi ? 3 : 1);
            v2f A, B;
            A.x = (lm == k0) ? 0.25f : 0.0f;       // column k of 0.25*I16
            A.y = (lm == k1) ? 0.25f : 0.0f;
            B.x = H[(long)k0 * DD + lm];           // rows k0,k1 of the H tile
            B.y = H[(long)k1 * DD + lm];
            acc = __builtin_amdgcn_wmma_f32_16x16x4_f32(
                false, A, false, B, (short)0, acc, false, false);
        }
    }

#pragma unroll
    for (int r = 0; r < 8; ++r) {
        int m = r + (hi ? 8 : 0);
        out[base + (long)m * DD + lm] = acc[r];
    }
}

// ---------------------------- host launcher -------------------------------
extern "C" void kernel_launch(void* const* d_in, const int* in_sizes, int n_in,
                              void* d_out, int out_size, void* d_ws, size_t ws_size,
                              hipStream_t stream)
{
    const int*   x       = (const int*)  d_in[0];
    const int*   esrc    = (const int*)  d_in[1];
    const int*   edst    = (const int*)  d_in[2];
    const float* ew      = (const float*)d_in[3];
    const float* embed_w = (const float*)d_in[4];
    float*       out     = (float*)d_out;

    const long nd = (long)NN * DD;                 // 3.2M floats per buffer
    if (ws_size < (size_t)(4 * nd * sizeof(float))) return;

    float* h0 = (float*)d_ws;
    float* h1 = h0 + nd;
    float* h2 = h1 + nd;
    float* h3 = h2 + nd;

    // h0 = embed_weight[x]
    gather_embed_kernel<<<(NN * (DD / 4)) / 256, 256, 0, stream>>>(x, embed_w, h0);

    // zero h1..h3 (scatter targets) — every call, so graph replays are clean
    zero_kernel<<<(int)((3 * nd / 4) / 256), 256, 0, stream>>>(
        (float4*)h1, (int)(3 * nd / 4));

    const int spmmGrid = (EE * 16) / 256;          // 50000

    // three fused dropout+SpMM layers (distinct seeds = distinct masks)
    spmm_dropout_kernel<<<spmmGrid, 256, 0, stream>>>(esrc, edst, ew, h0, h1, 0x1234567u);
    spmm_dropout_kernel<<<spmmGrid, 256, 0, stream>>>(esrc, edst, ew, h1, h2, 0x89abcdeu);
    spmm_dropout_kernel<<<spmmGrid, 256, 0, stream>>>(esrc, edst, ew, h2, h3, 0xf0f1f2fu);

    // out = (h0+h1+h2+h3)/4 on the matrix pipe
    const int ntiles = (NN / 16) * (DD / 16);      // 12500, 4 waves/block
    combine_wmma_kernel<<<ntiles / 4, 128, 0, stream>>>(h0, h1, h2, h3, out);
}